// Group_36764920054510
// MI455X (gfx1250) — compile-verified
//
#include <hip/hip_runtime.h>
#include <hip/hip_bf16.h>
#include <stdint.h>

typedef __attribute__((ext_vector_type(2))) float v2f;
typedef __attribute__((ext_vector_type(8))) float v8f;
typedef __attribute__((ext_vector_type(4))) unsigned int u32x4;
typedef __attribute__((ext_vector_type(8))) int i32x8;
typedef __attribute__((ext_vector_type(4))) int i32x4;

#define NPTS   8192
#define NGRP   512
#define KNN    32
#define CTILE  16      // centers per workgroup (one WMMA M dim)
#define CHUNK  512     // points per LDS d2 chunk
#define NCH    (NPTS / CHUNK)
#define BIGF   1e10f

// Output layout (floats): neigh_xyz | neigh_attr | center | center_attr
#define ONX  ((size_t)0)
#define ONA  ((size_t)(8*NGRP*KNN*3))              // 393216
#define OC   ((size_t)(2*8*NGRP*KNN*3))            // 786432
#define OCA  ((size_t)(2*8*NGRP*KNN*3 + 8*NGRP*3)) // 798720

#if __has_builtin(__builtin_amdgcn_tensor_load_to_lds) && __has_builtin(__builtin_amdgcn_s_wait_tensorcnt)
#define USE_TDM 1
#else
#define USE_TDM 0
#endif

// ---------------------------------------------------------------------------
// Kernel 1: farthest point sampling. One block per batch. Each of 256 threads
// register-caches its 32 points' coordinates AND running min-distance, so the
// 512 dependent iterations are pure VALU + a wave32 shuffle argmax + a single
// 8-entry cross-wave reduce (2 barriers/step instead of 9).
// ---------------------------------------------------------------------------
__global__ __launch_bounds__(256) void fps_kernel(const float* __restrict__ xyz,
                                                  float* __restrict__ out) {
    const int b = blockIdx.x;
    const int t = threadIdx.x;
    const int lane = t & 31, wave = t >> 5;
    __shared__ float wv[8];
    __shared__ int   wi[8];
    __shared__ int   curS;

    const float* base = xyz + (size_t)b * NPTS * 6;

    float px[32], py[32], pz[32], dists[32];
#pragma unroll
    for (int j = 0; j < 32; ++j) {
        const float* pp = base + (size_t)(j * 256 + t) * 6;
        px[j] = pp[0]; py[j] = pp[1]; pz[j] = pp[2];
        dists[j] = BIGF;
    }

    int cur = 0;
    for (int it = 0; it < NGRP; ++it) {
        const float* cp = base + (size_t)cur * 6;   // uniform address -> broadcast
        float cx = cp[0], cy = cp[1], cz = cp[2];
        if (t == 0) {
            size_t o = ((size_t)b * NGRP + it) * 3;
            out[OC  + o + 0] = cx;  out[OC  + o + 1] = cy;  out[OC  + o + 2] = cz;
            out[OCA + o + 0] = cp[3]; out[OCA + o + 1] = cp[4]; out[OCA + o + 2] = cp[5];
        }
        // register-resident min-distance update + thread-local argmax
        float bv = -1.0f; int bi = 0x7fffffff;
#pragma unroll
        for (int j = 0; j < 32; ++j) {
            float dx = px[j] - cx, dy = py[j] - cy, dz = pz[j] - cz;
            float d = fminf(dists[j], dx * dx + dy * dy + dz * dz);
            dists[j] = d;
            int p = j * 256 + t;
            if (d > bv || (d == bv && p < bi)) { bv = d; bi = p; }
        }
        // wave32 shuffle argmax (prefer larger value, tie -> smaller index)
        for (int off = 16; off > 0; off >>= 1) {
            float ov = __shfl_xor(bv, off, 32);
            int   oi = __shfl_xor(bi, off, 32);
            if (ov > bv || (ov == bv && oi < bi)) { bv = ov; bi = oi; }
        }
        if (lane == 0) { wv[wave] = bv; wi[wave] = bi; }
        __syncthreads();
        if (wave == 0) {
            float v = (lane < 8) ? wv[lane] : -2.0f;
            int   i = (lane < 8) ? wi[lane] : 0x7fffffff;
            for (int off = 4; off > 0; off >>= 1) {
                float ov = __shfl_xor(v, off, 32);
                int   oi = __shfl_xor(i, off, 32);
                if (ov > v || (ov == v && oi < i)) { v = ov; i = oi; }
            }
            if (lane == 0) curS = i;
        }
        __syncthreads();
        cur = curS;
    }
}

// ---------------------------------------------------------------------------
// Kernel 2: fused distance (WMMA f32 16x16x4) + top-32 selection + gather.
// Block = 256 threads (8 wave32), one block per (batch, 16-center tile).
// Points are staged chunk-by-chunk into double-buffered LDS by the Tensor
// Data Mover (wave 0 issues tensor_load_to_lds; overlapped with the previous
// chunk's selection phase; TENSORcnt + barrier for handoff).
// A fragment encodes (-2cx,-2cy,-2cz,|c|^2); B fragment is (x0,x1,x2,1);
// D tile + |x|^2 gives d2 in a 16x512 LDS tile, then each wave maintains the
// top-32 list for 2 centers via argmin rounds with early exit, followed by a
// shuffle rank-sort and direct gather/center-subtract to the output.
// ---------------------------------------------------------------------------
__global__ __launch_bounds__(256) void group_kernel(const float* __restrict__ xyz,
                                                    float* __restrict__ out) {
    const int b    = blockIdx.y;
    const int gt   = blockIdx.x;                 // 32 tiles of 16 centers
    const int t    = threadIdx.x;
    const int lane = t & 31;
    const int wave = t >> 5;
    const int half = lane >> 4;                  // 0: lanes 0-15, 1: lanes 16-31
    const int l16  = lane & 15;

    __shared__ float cX[CTILE], cY[CTILE], cZ[CTILE], cN2[CTILE];
    __shared__ float d2buf[CTILE][CHUNK];        // 32 KB
    __shared__ float bestD[CTILE * KNN];         // 2 KB
    __shared__ int   bestI[CTILE * KNN];         // 2 KB
#if USE_TDM
    __shared__ float pts[2][CHUNK * 6];          // 24 KB double-buffered stage
#endif

    const float* P = xyz + (size_t)b * NPTS * 6;

    if (t < CTILE) {
        int g = gt * CTILE + t;
        size_t o = OC + ((size_t)b * NGRP + g) * 3;
        float x = out[o], y = out[o + 1], z = out[o + 2];
        cX[t] = x; cY[t] = y; cZ[t] = z; cN2[t] = x * x + y * y + z * z;
    }
    for (int j = t; j < CTILE * KNN; j += 256) { bestD[j] = BIGF; bestI[j] = 0; }
    __syncthreads();

    // A fragment (16x4 f32): lanes 0-15 hold K0,K1; lanes 16-31 hold K2,K3.
    v2f afrag;
    afrag.x = half ? (-2.0f * cZ[l16]) : (-2.0f * cX[l16]);
    afrag.y = half ? (cN2[l16])        : (-2.0f * cY[l16]);

#if USE_TDM
    // TDM descriptor: 1-D tile of CHUNK*6 f32 (12 KB) from the point array.
    auto tdm_issue = [&](int ch) {
        uint64_t gaddr = (uint64_t)(const void*)(P + (size_t)ch * CHUNK * 6);
        uint32_t laddr = (uint32_t)(uintptr_t)(const void*)&pts[ch & 1][0];
        u32x4 g0 = { 1u,                                   // count=1, user desc
                     laddr,                                // lds_addr
                     (uint32_t)gaddr,                      // global_addr[31:0]
                     (uint32_t)((gaddr >> 32) & 0x1FFFFFFu) | (2u << 30) }; // [56:32] | type=2
        i32x8 g1 = { 0x20000,                              // data_size=4B
                     (int)0xC0000000,                      // tensor_dim0 = 49152 (lo16<<16)
                     0x10000,                              // tensor_dim1 = 1
                     (int)(3072u << 16),                   // tile_dim0 = CHUNK*6
                     0,                                    // tile_dim1/2 = 0
                     49152,                                // tensor_dim0_stride lo
                     0, 0 };
        i32x4 z4 = { 0, 0, 0, 0 };
#if defined(__clang_major__) && __clang_major__ >= 23
        i32x8 z8 = { 0, 0, 0, 0, 0, 0, 0, 0 };
        __builtin_amdgcn_tensor_load_to_lds(g0, g1, z4, z4, z8, 0);
#else
        __builtin_amdgcn_tensor_load_to_lds(g0, g1, z4, z4, 0);
#endif
    };
    if (wave == 0) tdm_issue(0);   // prologue: chunk 0 in flight
#endif

    for (int ch = 0; ch < NCH; ++ch) {
        const int cbase = ch * CHUNK;
#if USE_TDM
        if (wave == 0) __builtin_amdgcn_s_wait_tensorcnt(0);
        __syncthreads();                                // chunk ch staged in LDS
        if (wave == 0 && ch + 1 < NCH) tdm_issue(ch + 1);  // overlap next DMA
        const float* pb = &pts[ch & 1][0];
#endif
        // ---- compute phase: 32 point-tiles of 16, 4 tiles per wave ----
        for (int s = 0; s < (CHUNK / 16) / 8; ++s) {
            int col0 = (wave * 4 + s) * 16;
#if USE_TDM
            const float* pp = pb + (size_t)(col0 + l16) * 6;
#else
            const float* pp = P + (size_t)(cbase + col0 + l16) * 6;
            if (ch + 1 < NCH)
                __builtin_prefetch(pp + (size_t)CHUNK * 6, 0, 3);
#endif
            float x0 = pp[0], x1 = pp[1], x2 = pp[2];
            v2f bfrag;
            bfrag.x = half ? x2 : x0;
            bfrag.y = half ? 1.0f : x1;
            float xn2 = x0 * x0 + x1 * x1 + x2 * x2;
            v8f acc = {};
            acc = __builtin_amdgcn_wmma_f32_16x16x4_f32(
                /*neg_a=*/false, afrag, /*neg_b=*/false, bfrag,
                /*c_mod=*/(short)0, acc, /*reuse_a=*/false, /*reuse_b=*/false);
#pragma unroll
            for (int v = 0; v < 8; ++v) {
                int M = v + (half << 3);         // C/D layout: VGPR v, lane half
                d2buf[M][col0 + l16] = acc[v] + xn2;
            }
        }
        __syncthreads();
        // ---- selection phase: wave handles centers 2*wave, 2*wave+1 ----
        for (int mm = 0; mm < 2; ++mm) {
            const int m = wave * 2 + mm;
            for (int r = 0; r < KNN; ++r) {
                float mv = 3.0e38f; int mi = 0;
                for (int j = lane; j < CHUNK; j += 32) {
                    float d = d2buf[m][j];
                    if (d < mv || (d == mv && j < mi)) { mv = d; mi = j; }
                }
                for (int off = 16; off > 0; off >>= 1) {
                    float ov = __shfl_xor(mv, off, 32);
                    int   oi = __shfl_xor(mi, off, 32);
                    if (ov < mv || (ov == mv && oi < mi)) { mv = ov; mi = oi; }
                }
                float lv = bestD[m * KNN + lane]; int ll = lane;
                for (int off = 16; off > 0; off >>= 1) {
                    float ov = __shfl_xor(lv, off, 32);
                    int   oi = __shfl_xor(ll, off, 32);
                    if (ov > lv || (ov == lv && oi > ll)) { lv = ov; ll = oi; }
                }
                if (!(mv < lv)) break;           // chunk can no longer improve list
                if (lane == 0) {
                    bestD[m * KNN + ll] = mv;
                    bestI[m * KNN + ll] = cbase + mi;
                    d2buf[m][mi] = 3.0e38f;
                }
            }
        }
        __syncthreads();
    }

    // ---- rank-sort (ascending d2, ties by point index) + gather/output ----
    for (int mm = 0; mm < 2; ++mm) {
        const int m = wave * 2 + mm;
        float v  = bestD[m * KNN + lane];
        int   pi = bestI[m * KNN + lane];
        int rank = 0;
        for (int j = 0; j < 32; ++j) {
            float vj = __shfl(v, j, 32);
            int   ij = __shfl(pi, j, 32);
            if (vj < v || (vj == v && ij < pi)) ++rank;
        }
        const int g = gt * CTILE + m;
        const float* pp = P + (size_t)pi * 6;
        size_t base3 = (((size_t)b * NGRP + g) * KNN + rank) * 3;
        out[ONX + base3 + 0] = pp[0] - cX[m];
        out[ONX + base3 + 1] = pp[1] - cY[m];
        out[ONX + base3 + 2] = pp[2] - cZ[m];
        out[ONA + base3 + 0] = pp[3];
        out[ONA + base3 + 1] = pp[4];
        out[ONA + base3 + 2] = pp[5];
    }
}

extern "C" void kernel_launch(void* const* d_in, const int* in_sizes, int n_in,
                              void* d_out, int out_size, void* d_ws, size_t ws_size,
                              hipStream_t stream) {
    (void)in_sizes; (void)n_in; (void)out_size; (void)d_ws; (void)ws_size;
    const float* xyz = (const float*)d_in[0];
    float* out = (float*)d_out;
    // Kernel 1 writes center/center_attr into d_out; kernel 2 reads them
    // (same stream -> ordered).
    fps_kernel<<<dim3(8), dim3(256), 0, stream>>>(xyz, out);
    group_kernel<<<dim3(32, 8), dim3(256), 0, stream>>>(xyz, out);
}